// YOLOLoss_21105469292946
// MI455X (gfx1250) — compile-verified
//
#include <hip/hip_runtime.h>
#include <hip/hip_bf16.h>
#include <math.h>

typedef __attribute__((ext_vector_type(2))) float v2f;
typedef __attribute__((ext_vector_type(8))) float v8f;

#define BB   64
#define TT   50
#define GH   64
#define GW   64
#define HWC  4096      // GH*GW
#define NCLS 80
#define CH   85        // 5 + NCLS
#define NTGT (BB*TT)   // 3200

__device__ __forceinline__ float softplusf(float x) {
    // stable softplus: max(x,0) + log1p(exp(-|x|))
    return fmaxf(x, 0.0f) + log1pf(expf(-fabsf(x)));
}

// Exact full-wave (32-lane) f32 sum using V_WMMA_F32_16X16X4_F32.
// A (16x4): lane m in [0,16): A[m][0]=v, A[m][1]=0 ; lane m+16: A[m][2]=v, A[m][3]=0.
// B = ones(4x16)  =>  D[m][n] = v[m] + v[m+16] for all n.
// D VGPR j: lanes 0-15 hold M=j, lanes 16-31 hold M=j+8.
// sum_j d[j] gives half-sums in lane groups; xor-16 shuffle completes the sum.
// Requires EXEC all-ones: call only from fully-active, non-divergent code.
__device__ __forceinline__ float wave_sum_wmma(float v) {
    v2f a; a.x = v;     a.y = 0.0f;
    v2f b; b.x = 1.0f;  b.y = 1.0f;
    v8f c = {};
    v8f d = __builtin_amdgcn_wmma_f32_16x16x4_f32(
        /*neg_a=*/false, a, /*neg_b=*/false, b,
        /*c_mod=*/(short)0, c, /*reuse_a=*/false, /*reuse_b=*/false);
    float s = d[0] + d[1] + d[2] + d[3] + d[4] + d[5] + d[6] + d[7];
    s += __shfl_xor(s, 16, 32);
    return s;   // full wave sum, replicated in every lane
}

// ---------------------------------------------------------------- kernel 1
__global__ void yolo_init_kernel(unsigned int* __restrict__ mask,
                                 float* __restrict__ img_noobj,
                                 float* __restrict__ tgt_sum) {
    int i = blockIdx.x * 256 + threadIdx.x;
    if (i < BB * 128) mask[i] = 0u;          // 64 images x 4096 bits
    if (i < BB)       img_noobj[i] = 0.0f;
    if (i == 0)       tgt_sum[0] = 0.0f;
}

// ---------------------------------------------------------------- kernel 2
// one wave per target; 8 targets per 256-thread block
__global__ void yolo_target_kernel(const float* __restrict__ pred,
                                   const float* __restrict__ tbox,
                                   const int*   __restrict__ tcls,
                                   unsigned int* __restrict__ mask,
                                   float* __restrict__ tgt_sum) {
    const int lane = threadIdx.x & 31;
    const int t    = blockIdx.x * 8 + (threadIdx.x >> 5);   // < 3200 exactly
    const int b    = t / TT;
    const int j    = t - b * TT;

    const float* box = tbox + ((size_t)b * TT + j) * 4;
    const float cx = box[0], cy = box[1], w = box[2], h = box[3];
    const float gxf = floorf(cx * (float)GW);
    const float gyf = floorf(cy * (float)GH);
    const int   gi  = (int)(gyf * (float)GW + gxf);

    const float* p = pred + ((size_t)b * HWC + (size_t)gi) * CH;

    // class log-sum-exp over 80 logits, 32 lanes x up to 3 elements
    const int k0 = lane, k1 = lane + 32, k2 = lane + 64;
    const bool ok2 = (k2 < NCLS);
    const float v0 = p[5 + k0];
    const float v1 = p[5 + k1];
    const float v2 = ok2 ? p[5 + k2] : 0.0f;

    float m = fmaxf(fmaxf(v0, v1), ok2 ? v2 : -INFINITY);
    #pragma unroll
    for (int off = 16; off; off >>= 1) m = fmaxf(m, __shfl_xor(m, off, 32));

    const int c = tcls[b * TT + j];
    float s    = expf(v0 - m) + expf(v1 - m) + (ok2 ? expf(v2 - m) : 0.0f);
    float pick = (k0 == c ? v0 : 0.0f) + (k1 == c ? v1 : 0.0f)
               + ((ok2 && k2 == c) ? v2 : 0.0f);
    #pragma unroll
    for (int off = 16; off; off >>= 1) {
        s    += __shfl_xor(s,    off, 32);
        pick += __shfl_xor(pick, off, 32);
    }

    if (lane == 0) {
        const float cls_loss = (m + logf(s)) - pick;

        const float tx = cx * (float)GW - gxf;
        const float ty = cy * (float)GH - gyf;
        const float sx = 1.0f / (1.0f + expf(-p[0]));
        const float sy = 1.0f / (1.0f + expf(-p[1]));
        const float xy = 0.5f * ((sx - tx) * (sx - tx) + (sy - ty) * (sy - ty));

        const float tw = logf(w * (float)GW + 1e-16f);
        const float th = logf(h * (float)GH + 1e-16f);
        const float wh = 0.5f * ((p[2] - tw) * (p[2] - tw) + (p[3] - th) * (p[3] - th));

        const float conf = softplusf(-p[4]);

        const float per = 5.0f * (xy + wh) + conf + cls_loss;
        atomicAdd(tgt_sum, per);
        atomicOr(&mask[b * 128 + (gi >> 5)], 1u << (gi & 31));
    }
}

// ---------------------------------------------------------------- kernel 3
// 1024 blocks: blockIdx = b*16 + chunk; 256 cells per block
__global__ void yolo_noobj_kernel(const float* __restrict__ pred,
                                  const unsigned int* __restrict__ mask,
                                  float* __restrict__ img_noobj) {
    const int b     = (int)(blockIdx.x >> 4);
    const int chunk = (int)(blockIdx.x & 15);
    const int hw    = chunk * 256 + threadIdx.x;

    const float conf = pred[((size_t)b * HWC + (size_t)hw) * CH + 4];
    const unsigned int mword = mask[b * 128 + (hw >> 5)];
    const float val = ((mword >> (hw & 31)) & 1u) ? 0.0f : softplusf(conf);

    // full-wave WMMA reduction (EXEC all-ones: no divergence above)
    const float wsum = wave_sum_wmma(val);

    __shared__ float lds[8];
    const int lane = threadIdx.x & 31;
    const int wave = threadIdx.x >> 5;
    if (lane == 0) lds[wave] = wsum;
    __syncthreads();
    if (threadIdx.x == 0) {
        float bs = 0.0f;
        #pragma unroll
        for (int i = 0; i < 8; ++i) bs += lds[i];
        atomicAdd(&img_noobj[b], bs);
    }
}

// ---------------------------------------------------------------- kernel 4
__global__ void yolo_final_kernel(const unsigned int* __restrict__ mask,
                                  const float* __restrict__ img_noobj,
                                  const float* __restrict__ tgt_sum,
                                  float* __restrict__ out) {
    const int b = threadIdx.x;          // blockDim.x == 64, one thread per image
    int cnt = 0;
    #pragma unroll 4
    for (int i = 0; i < 128; ++i) cnt += __popc(mask[b * 128 + i]);
    const float noobj_cnt = (float)(HWC - cnt);
    float contrib = img_noobj[b] / fmaxf(noobj_cnt, 1.0f);

    #pragma unroll
    for (int off = 16; off; off >>= 1) contrib += __shfl_xor(contrib, off, 32);

    __shared__ float lds2[2];
    if ((threadIdx.x & 31) == 0) lds2[threadIdx.x >> 5] = contrib;
    __syncthreads();
    if (threadIdx.x == 0) {
        const float noobj_loss = lds2[0] + lds2[1];
        out[0] = (tgt_sum[0] + 0.5f * noobj_loss) / (float)NTGT;
    }
}

// ---------------------------------------------------------------- launch
extern "C" void kernel_launch(void* const* d_in, const int* in_sizes, int n_in,
                              void* d_out, int out_size, void* d_ws, size_t ws_size,
                              hipStream_t stream) {
    const float* pred = (const float*)d_in[0];   // [64, 4096, 85] f32
    const float* tbox = (const float*)d_in[1];   // [64, 50, 4]    f32
    const int*   tcls = (const int*)d_in[2];     // [64, 50]       i32
    // d_in[3], d_in[4] = grid_h, grid_w scalars; fixed at 64 (compile-time).

    unsigned int* mask      = (unsigned int*)d_ws;                 // 8192 u32 = 32 KB
    float*        img_noobj = (float*)((char*)d_ws + 8192 * 4);    // 64 f32
    float*        tgt_sum   = img_noobj + 64;                      // 1 f32

    float* outf = (float*)d_out;

    yolo_init_kernel  <<<32,   256, 0, stream>>>(mask, img_noobj, tgt_sum);
    yolo_target_kernel<<<NTGT / 8, 256, 0, stream>>>(pred, tbox, tcls, mask, tgt_sum);
    yolo_noobj_kernel <<<BB * 16, 256, 0, stream>>>(pred, mask, img_noobj);
    yolo_final_kernel <<<1,    64,  0, stream>>>(mask, img_noobj, tgt_sum, outf);
}